// CategoricalMap2D_18992345383120
// MI455X (gfx1250) — compile-verified
//
#include <hip/hip_runtime.h>

// ---------------- problem constants ----------------
#define BB 4
#define TT 8
#define HH 120
#define WW 160
#define NSEM 4
#define CAT 16
#define MAPC 20      // NON_SEM + CAT
#define FEATC 24     // 2*NON_SEM + CAT
#define LM 240
#define GM 960
#define M2 (LM*LM)   // 57600
#define G2 (GM*GM)   // 921600
#define HW (HH*WW)   // 19200
#define PI_F 3.14159265358979323846f
// FX = (W/2) / tan(deg2rad(79)/2) = 97.0477 ; INV_FX = tan(39.5deg)/80
#define INV_FX 0.0103042116f

#define AS1 __attribute__((address_space(1)))
#define AS3 __attribute__((address_space(3)))

typedef __attribute__((ext_vector_type(2))) float v2f;
typedef __attribute__((ext_vector_type(8))) float v8f;

// state layout in ws (after splat grids), as floats:
//  [0..11]  lpose  (B x 3)
//  [12..23] gpose  (B x 3)
//  [24..35] origins(B x 3)
//  int view [36..51] bounds (B x 4)
//  [52..59] (cos th, sin th) per batch, post-integration

__global__ void k_init_state(const float* lp0, const float* gp0, const int* bd0,
                             const float* og0, float* st) {
    int i = threadIdx.x;
    if (i < 12) { st[i] = lp0[i]; st[12 + i] = gp0[i]; st[24 + i] = og0[i]; }
    if (i < 16) ((int*)st)[36 + i] = bd0[i];
}

__global__ void k_reset_gmap(float* gmap, const int* done, int t) {
    size_t idx = (size_t)blockIdx.x * blockDim.x + threadIdx.x;
    if (idx >= (size_t)BB * MAPC * G2) return;
    int b = (int)(idx / ((size_t)MAPC * G2));
    if (done[b * TT + t]) gmap[idx] = 0.0f;
}

__global__ void k_reset_lmap(float* lmap, const int* done, int t) {
    size_t idx = (size_t)blockIdx.x * blockDim.x + threadIdx.x;
    if (idx >= (size_t)BB * MAPC * M2) return;
    int b = (int)(idx / ((size_t)MAPC * M2));
    if (done[b * TT + t]) lmap[idx] = 0.0f;
}

__global__ void k_pose(const float* delta, const int* done, const int* upd,
                       float* st, int t,
                       float* lp_o, float* gp_o, float* bn_o, float* og_o) {
    int b = threadIdx.x;
    if (b >= BB) return;
    float* lp = st + 3 * b;
    float* gp = st + 12 + 3 * b;
    float* og = st + 24 + 3 * b;
    int* bd = ((int*)st) + 36 + 4 * b;
    if (done[b * TT + t]) {
        lp[0] = 600.0f; lp[1] = 600.0f; lp[2] = 0.0f;          // LOCAL_POSE0
        gp[0] = 2400.0f; gp[1] = 2400.0f; gp[2] = 0.0f;        // GLOBAL_POSE0
        og[0] = 1800.0f; og[1] = 1800.0f; og[2] = 0.0f;        // ORIGINS0
        bd[0] = 360; bd[1] = 600; bd[2] = 360; bd[3] = 600;    // BOUNDS0
    }
    const float* d = delta + (size_t)(b * TT + t) * 3;
    float th = lp[2] * (PI_F / 180.0f);
    float c = cosf(th), s = sinf(th);
    float dx = d[0] * c - d[1] * s;
    float dy = d[0] * s + d[1] * c;
    lp[0] += dx; lp[1] += dy; lp[2] += d[2];
    // post-integration heading sincos for the splat kernel (per-batch scalar)
    float th2 = lp[2] * (PI_F / 180.0f);
    st[52 + 2 * b] = cosf(th2);
    st[53 + 2 * b] = sinf(th2);
    if (upd[b * TT + t]) {
        gp[0] = lp[0] + og[0]; gp[1] = lp[1] + og[1]; gp[2] = lp[2] + og[2];
    }
    size_t o3 = (size_t)(b * TT + t) * 3;
    lp_o[o3 + 0] = lp[0]; lp_o[o3 + 1] = lp[1]; lp_o[o3 + 2] = lp[2];
    gp_o[o3 + 0] = gp[0]; gp_o[o3 + 1] = gp[1]; gp_o[o3 + 2] = gp[2];
    og_o[o3 + 0] = og[0]; og_o[o3 + 1] = og[1]; og_o[o3 + 2] = og[2];
    size_t o4 = (size_t)(b * TT + t) * 4;
    bn_o[o4 + 0] = (float)bd[0]; bn_o[o4 + 1] = (float)bd[1];
    bn_o[o4 + 2] = (float)bd[2]; bn_o[o4 + 3] = (float)bd[3];
}

__global__ void k_zero_splat(float* splat) {
    size_t idx = (size_t)blockIdx.x * blockDim.x + threadIdx.x;
    if (idx < (size_t)BB * 18 * M2) splat[idx] = 0.0f;
}

// Backproject + 4x4 camera transform via V_WMMA_F32_16X16X4_F32 (16 points / wave)
// then scatter-add into 18-channel splat grids.
__global__ __launch_bounds__(256) void k_splat(const float* __restrict__ obs,
                                               const float* __restrict__ camT,
                                               const float* __restrict__ st,
                                               float* __restrict__ splat, int t) {
    __shared__ float lds[8][16][16];   // [wave][out-row i][point m]
    int lane    = threadIdx.x & 31;
    int wave    = threadIdx.x >> 5;
    int lane_lo = lane & 15;
    int hi      = lane >> 4;           // 0: K=0,1 ; 1: K=2,3
    int chunk   = blockIdx.x * 8 + wave;       // 4800 chunks of 16 points
    int pglob   = chunk * 16 + lane_lo;        // < B*H*W = 76800
    int b       = pglob / HW;
    int pix     = pglob % HW;
    int u       = pix % WW;
    int v       = pix / WW;

    const float* ob = obs + (size_t)(b * TT + t) * MAPC * HW;
    float depth = ob[3 * HW + pix] * 400.0f + 50.0f;
    float X = ((float)u - WW * 0.5f) * depth * INV_FX;
    float Y = (HH * 0.5f - (float)v) * depth * INV_FX;

    // A: 16x4 (MxK). lanes 0-15 hold K=0,1 ; lanes 16-31 hold K=2,3 of point m=lane_lo
    v2f a;
    a.x = hi ? depth : X;
    a.y = hi ? 1.0f  : Y;
    // B: 4x16 (KxN). column n holds cam_T^T, i.e. B[k][n] = T[n][k]; cols >=4 are zero
    const float* T = camT + (size_t)(b * TT + t) * 16;
    v2f bm;
    bm.x = (lane_lo < 4) ? T[lane_lo * 4 + 2 * hi]     : 0.0f;
    bm.y = (lane_lo < 4) ? T[lane_lo * 4 + 2 * hi + 1] : 0.0f;

    v8f acc = {};
    acc = __builtin_amdgcn_wmma_f32_16x16x4_f32(
        /*neg_a=*/false, a, /*neg_b=*/false, bm,
        /*c_mod=*/(short)0, acc, /*reuse_a=*/false, /*reuse_b=*/false);

    // D layout: VGPR r, lanes 0-15: (M=r, N=lane); lanes 16-31: (M=r+8, N=lane-16)
    if (lane_lo < 4) {
        int n = lane_lo, mbase = hi * 8;
#pragma unroll
        for (int r = 0; r < 8; r++) lds[wave][n][mbase + r] = acc[r];
    }
    __syncthreads();

    float p0 = lds[wave][0][lane_lo];   // pts'[0]
    float p1 = lds[wave][1][lane_lo];   // pts'[1]
    float p2 = lds[wave][2][lane_lo];   // pts'[2]
    float x_f = p2;
    float y_l = -p0;
    float z_u = p1 + 88.0f;

    float cth = st[52 + 2 * b];
    float sth = st[53 + 2 * b];
    float wx = st[3 * b + 0] + x_f * cth - y_l * sth;
    float wy = st[3 * b + 1] + x_f * sth + y_l * cth;
    int ix = (int)floorf(wx * 0.2f);
    int iy = (int)floorf(wy * 0.2f);
    bool valid = (depth > 25.0f) && (depth < 500.0f) &&
                 (ix >= 0) && (ix < LM) && (iy >= 0) && (iy < LM);
    int cix = min(max(ix, 0), LM - 1);
    int ciy = min(max(iy, 0), LM - 1);
    int flat = ciy * LM + cix;

    float* sg = splat + (size_t)b * 18 * M2;
    if (hi == 0 && valid) {
        atomicAdd(&sg[0 * M2 + flat], 1.0f);                      // exp
        if (z_u > 25.0f && z_u < 150.0f)
            atomicAdd(&sg[1 * M2 + flat], 1.0f);                  // obstacle
    }
    if (valid) {
        int c0 = hi * 8;                                          // split 16 sem channels
#pragma unroll
        for (int cc = 0; cc < 8; cc++) {
            float sv = ob[(NSEM + c0 + cc) * HW + pix];
            atomicAdd(&sg[(2 + c0 + cc) * M2 + flat], sv);
        }
    }
}

__global__ void k_fuse_local(const float* __restrict__ splat,
                             const float* __restrict__ st,
                             float* __restrict__ lmap) {
    int idx = blockIdx.x * blockDim.x + threadIdx.x;
    if (idx >= BB * M2) return;
    int b = idx / M2, cell = idx % M2;
    int x = cell % LM, y = cell / LM;
    const float* lp = st + 3 * b;
    float dxg = (float)x - lp[0] * 0.2f;
    float dyg = (float)y - lp[1] * 0.2f;
    float d2 = dxg * dxg + dyg * dyg;
    float agent = (d2 <= 4.0f) ? 1.0f : 0.0f;
    float close = (d2 <= 9.0f) ? 1.0f : 0.0f;
    const float* sg = splat + (size_t)b * 18 * M2;
    float* lm = lmap + (size_t)b * MAPC * M2;
    float obst = fminf(fmaxf(sg[1 * M2 + cell], 0.0f), 1.0f);
    float expg = fminf(fmaxf(sg[0 * M2 + cell], 0.0f), 1.0f);
    lm[0 * M2 + cell] = fmaxf(lm[0 * M2 + cell], obst);
    lm[1 * M2 + cell] = fmaxf(lm[1 * M2 + cell], expg);
    lm[2 * M2 + cell] = agent;
    lm[3 * M2 + cell] = fmaxf(lm[3 * M2 + cell], close);
#pragma unroll
    for (int c = 0; c < CAT; c++) {
        float s = fminf(fmaxf(sg[(2 + c) * M2 + cell], 0.0f), 1.0f);
        lm[(NSEM + c) * M2 + cell] = fmaxf(lm[(NSEM + c) * M2 + cell], s);
    }
}

__global__ __launch_bounds__(256) void k_fuse_global(const float* __restrict__ lmap,
                                                     float* __restrict__ gmap,
                                                     const float* __restrict__ st,
                                                     const int* __restrict__ upd,
                                                     float* __restrict__ feats, int t) {
    __shared__ float stage[256];
    size_t idx = (size_t)blockIdx.x * blockDim.x + threadIdx.x;
    if (idx >= (size_t)BB * MAPC * M2) return;
    int b = (int)(idx / ((size_t)MAPC * M2));
    int r = (int)(idx % ((size_t)MAPC * M2));
    int c = r / M2, cell = r % M2;
    int y = cell / LM, x = cell % LM;
    const int* bd = ((const int*)st) + 36 + 4 * b;
    int y1 = bd[0], x1 = bd[2];
    size_t gi = ((size_t)(b * MAPC + c) * GM + (size_t)(y1 + y)) * GM + (size_t)(x1 + x);

    float crop;
#if defined(__gfx1250__) && \
    __has_builtin(__builtin_amdgcn_global_load_async_to_lds_b32) && \
    __has_builtin(__builtin_amdgcn_s_wait_asynccnt)
    // async DMA the crop element into this lane's LDS slot (ASYNCcnt-tracked)
    __builtin_amdgcn_global_load_async_to_lds_b32(
        (AS1 int*)(gmap + gi), (AS3 int*)&stage[threadIdx.x], 0, 0);
    __builtin_amdgcn_s_wait_asynccnt(0);
    crop = stage[threadIdx.x];
#else
    crop = gmap[gi];
    stage[threadIdx.x] = crop;   // keep LDS alive either way
#endif

    float l = lmap[(size_t)(b * MAPC + c) * M2 + cell];
    float fused = upd[b * TT + t] ? fmaxf(crop, l) : crop;
    gmap[gi] = fused;
    float* fo = feats + (size_t)(b * TT + t) * FEATC * M2;
    if (c < NSEM) {
        fo[c * M2 + cell] = l;                 // lmap[:, :4]
        fo[(NSEM + c) * M2 + cell] = fused;    // fused[:, :4]
    } else {
        fo[(NSEM + c) * M2 + cell] = l;        // lmap[:, 4:]
    }
}

extern "C" void kernel_launch(void* const* d_in, const int* in_sizes, int n_in,
                              void* d_out, int out_size, void* d_ws, size_t ws_size,
                              hipStream_t stream) {
    const float* obs   = (const float*)d_in[0];
    const float* delta = (const float*)d_in[1];
    const int*   done  = (const int*)d_in[2];
    const int*   upd   = (const int*)d_in[3];
    const float* camT  = (const float*)d_in[4];
    const float* lmap0 = (const float*)d_in[5];
    const float* gmap0 = (const float*)d_in[6];
    const float* lp0   = (const float*)d_in[7];
    const float* gp0   = (const float*)d_in[8];
    const int*   bd0   = (const int*)d_in[9];
    const float* og0   = (const float*)d_in[10];

    float* out   = (float*)d_out;
    float* feats = out;
    float* lmap  = feats + (size_t)BB * TT * FEATC * M2;   // (B,20,240,240) carry
    float* gmap  = lmap + (size_t)BB * MAPC * M2;          // (B,20,960,960) carry
    float* lp_o  = gmap + (size_t)BB * MAPC * G2;
    float* gp_o  = lp_o + (size_t)BB * TT * 3;
    float* bn_o  = gp_o + (size_t)BB * TT * 3;
    float* og_o  = bn_o + (size_t)BB * TT * 4;

    float* splat = (float*)d_ws;                           // (B,18,240,240)
    float* st    = splat + (size_t)BB * 18 * M2;           // pose/bounds state

    // seed scan carry from inputs (device-to-device, capture-safe)
    (void)hipMemcpyAsync(lmap, lmap0, sizeof(float) * (size_t)BB * MAPC * M2,
                         hipMemcpyDeviceToDevice, stream);
    (void)hipMemcpyAsync(gmap, gmap0, sizeof(float) * (size_t)BB * MAPC * G2,
                         hipMemcpyDeviceToDevice, stream);
    k_init_state<<<1, 32, 0, stream>>>(lp0, gp0, bd0, og0, st);

    const int TH = 256;
    size_t n_g = (size_t)BB * MAPC * G2;
    size_t n_l = (size_t)BB * MAPC * M2;
    size_t n_s = (size_t)BB * 18 * M2;

    for (int t = 0; t < TT; t++) {
        k_reset_gmap<<<(unsigned)((n_g + TH - 1) / TH), TH, 0, stream>>>(gmap, done, t);
        k_reset_lmap<<<(unsigned)((n_l + TH - 1) / TH), TH, 0, stream>>>(lmap, done, t);
        k_pose<<<1, 32, 0, stream>>>(delta, done, upd, st, t, lp_o, gp_o, bn_o, og_o);
        k_zero_splat<<<(unsigned)((n_s + TH - 1) / TH), TH, 0, stream>>>(splat);
        // 76800 points / 16 per wave = 4800 waves; 8 waves/block -> 600 blocks
        k_splat<<<600, 256, 0, stream>>>(obs, camT, st, splat, t);
        k_fuse_local<<<(BB * M2 + TH - 1) / TH, TH, 0, stream>>>(splat, st, lmap);
        k_fuse_global<<<(unsigned)((n_l + TH - 1) / TH), TH, 0, stream>>>(lmap, gmap, st, upd, feats, t);
    }
}